// JKNetMaxpool_87600152969921
// MI455X (gfx1250) — compile-verified
//
#include <hip/hip_runtime.h>
#include <hip/hip_bf16.h>

typedef __attribute__((ext_vector_type(2))) float v2f;
typedef __attribute__((ext_vector_type(8))) float v8f;

#define H_DIM 128

__device__ __forceinline__ void atomic_add_f32(float* p, float v) {
  __hip_atomic_fetch_add(p, v, __ATOMIC_RELAXED, __HIP_MEMORY_SCOPE_AGENT);
}

// ---------------------------------------------------------------------------
// Degree counting: one thread per edge, fp32 atomics into out/in degree arrays
// ---------------------------------------------------------------------------
__global__ __launch_bounds__(256) void degree_kernel(const int* __restrict__ src,
                                                     const int* __restrict__ dst,
                                                     float* __restrict__ odeg,
                                                     float* __restrict__ ideg, int E) {
  int i = blockIdx.x * 256 + threadIdx.x;
  if (i < E) {
    atomic_add_f32(&odeg[src[i]], 1.0f);
    atomic_add_f32(&ideg[dst[i]], 1.0f);
  }
}

__global__ __launch_bounds__(256) void norm_kernel(const float* __restrict__ odeg,
                                                   const float* __restrict__ ideg,
                                                   float* __restrict__ ns,
                                                   float* __restrict__ nd, int N) {
  int i = blockIdx.x * 256 + threadIdx.x;
  if (i < N) {
    float o = odeg[i], d = ideg[i];
    ns[i] = o > 0.f ? rsqrtf(o) : 0.f;
    nd[i] = d > 0.f ? rsqrtf(d) : 0.f;
  }
}

// ---------------------------------------------------------------------------
// Weight transpose: Wt[c][k] = W[k][c] (tiny, once/launch) so the GEMM
// B-fragment is one contiguous global_load_b64 per lane.
// ---------------------------------------------------------------------------
__global__ __launch_bounds__(256) void transpose_kernel(const float* __restrict__ W,
                                                        float* __restrict__ Wt,
                                                        int K, int C) {
  int i = blockIdx.x * 256 + threadIdx.x;
  if (i < K * C) {
    int k = i / C;
    int c = i % C;
    Wt[(size_t)c * K + k] = W[i];
  }
}

// ---------------------------------------------------------------------------
// P[N,128] = (A[N,K] @ B[K,128]) * ns[row],  B TRANSPOSED: Bt[128][K].
// Block: 256 threads = 8 waves, covers MT*16 rows x 128 cols.
// Wave w owns cols [16w,16w+16) and MT row-tiles -> each B fragment is reused
// for MT WMMAs (4x less B traffic from L2 at MT=4).
// A tile (MT*16 x K) staged in LDS, +4 pad => stride = 4 dwords mod 64 banks,
// so the 32 lanes' b64 reads are bank-conflict free.
//   A frag: lane(M = lane&15) holds K = k0 + (lane>>4)*2 + {0,1}
//   B frag: lane(N = lane&15) holds K = k0 + (lane>>4)*2 + {0,1}
//   C/D:    elem j -> row M = j + (lane>=16 ? 8 : 0), col N = lane&15
// ---------------------------------------------------------------------------
template <int K, int MT>
__global__ __launch_bounds__(256) void gemm_scale_kernel(
    const float* __restrict__ A, const float* __restrict__ Bt,
    const float* __restrict__ ns, float* __restrict__ P, int N) {
  __shared__ float As[MT * 16][K + 4];
  const int tid  = threadIdx.x;
  const int wave = tid >> 5;
  const int lane = tid & 31;
  const int row0 = blockIdx.x * (MT * 16);

  // cooperative, coalesced float4 load of the (MT*16) x K A tile
  for (int i = tid; i < MT * 16 * (K / 4); i += 256) {
    int r  = i / (K / 4);
    int c4 = i % (K / 4);
    int rr = row0 + r;
    if (rr >= N) rr = N - 1;  // clamp (harmless duplicate read)
    float4 v = reinterpret_cast<const float4*>(A + (size_t)rr * K)[c4];
    As[r][c4 * 4 + 0] = v.x;
    As[r][c4 * 4 + 1] = v.y;
    As[r][c4 * 4 + 2] = v.z;
    As[r][c4 * 4 + 3] = v.w;
  }
  __syncthreads();

  const int mn   = lane & 15;
  const int koff = (lane >> 4) << 1;  // 0 or 2
  const int col  = wave * 16 + mn;
  const float* Btc = Bt + (size_t)col * K + koff;   // contiguous in k

  v8f acc[MT] = {};
#pragma unroll 4
  for (int k0 = 0; k0 < K; k0 += 4) {
    v2f b = *reinterpret_cast<const v2f*>(Btc + k0);  // global_load_b64
#pragma unroll
    for (int m = 0; m < MT; ++m) {
      v2f a = *reinterpret_cast<const v2f*>(&As[m * 16 + mn][k0 + koff]);
      acc[m] = __builtin_amdgcn_wmma_f32_16x16x4_f32(false, a, false, b, (short)0,
                                                     acc[m], false, false);
    }
  }

  const int sub = (lane >> 4) << 3;  // 0 or 8
  if (row0 + MT * 16 <= N) {
    // fast path: branch-free pointer-increment stores
#pragma unroll
    for (int m = 0; m < MT; ++m) {
      const int rbase = row0 + m * 16 + sub;
      float* p = P + (size_t)rbase * H_DIM + col;
      const float* nsr = ns + rbase;
#pragma unroll
      for (int j = 0; j < 8; ++j) {
        *p = acc[m][j] * nsr[j];
        p += H_DIM;
      }
    }
  } else {
#pragma unroll
    for (int m = 0; m < MT; ++m) {
      const int rbase = row0 + m * 16 + sub;
#pragma unroll
      for (int j = 0; j < 8; ++j) {
        int r = rbase + j;
        if (r < N) P[(size_t)r * H_DIM + col] = acc[m][j] * ns[r];
      }
    }
  }
}

// ---------------------------------------------------------------------------
// out[N,C] = A[N,128] @ W[128,C] + bias ; C=40 padded to 48 cols in LDS so the
// K-loop is branch-free. Block: 96 threads = 3 waves, one 16-col tile each.
// ---------------------------------------------------------------------------
__global__ __launch_bounds__(96) void gemm_final_kernel(
    const float* __restrict__ A, const float* __restrict__ W,
    const float* __restrict__ bias, float* __restrict__ out, int N, int C) {
  __shared__ float As[16][H_DIM + 4];
  __shared__ float Wt[48][H_DIM + 4];   // transposed + zero padded
  const int tid  = threadIdx.x;
  const int wave = tid >> 5;
  const int lane = tid & 31;
  const int row0 = blockIdx.x << 4;

  for (int i = tid; i < 16 * (H_DIM / 4); i += 96) {
    int r  = i >> 5;   // H_DIM/4 == 32 float4 per row
    int c4 = i & 31;
    int rr = row0 + r;
    if (rr >= N) rr = N - 1;
    float4 v = reinterpret_cast<const float4*>(A + (size_t)rr * H_DIM)[c4];
    As[r][c4 * 4 + 0] = v.x;
    As[r][c4 * 4 + 1] = v.y;
    As[r][c4 * 4 + 2] = v.z;
    As[r][c4 * 4 + 3] = v.w;
  }
  for (int i = tid; i < 48 * H_DIM; i += 96) {
    int c = i / H_DIM;
    int k = i % H_DIM;
    Wt[c][k] = (c < C) ? W[(size_t)k * C + c] : 0.f;
  }
  __syncthreads();

  const int mn   = lane & 15;
  const int koff = (lane >> 4) << 1;
  const int col  = wave * 16 + mn;
  v8f acc = {};
#pragma unroll 8
  for (int k0 = 0; k0 < H_DIM; k0 += 4) {
    v2f a = *reinterpret_cast<const v2f*>(&As[mn][k0 + koff]);
    v2f b = *reinterpret_cast<const v2f*>(&Wt[col][k0 + koff]);
    acc = __builtin_amdgcn_wmma_f32_16x16x4_f32(false, a, false, b, (short)0,
                                                acc, false, false);
  }

  if (col < C) {
    const int   rbase = row0 + ((lane >> 4) << 3);
    const float bb    = bias[col];
    if (row0 + 16 <= N) {
      float* o = out + (size_t)rbase * C + col;
#pragma unroll
      for (int j = 0; j < 8; ++j) {
        *o = acc[j] + bb;
        o += C;
      }
    } else {
#pragma unroll
      for (int j = 0; j < 8; ++j) {
        int r = rbase + j;
        if (r < N) out[(size_t)r * C + col] = acc[j] + bb;
      }
    }
  }
}

// ---------------------------------------------------------------------------
// Edge aggregation: one wave per edge. Lanes read src row as float4 (coalesced
// 512B) and scatter 4 fp32 atomic adds into the dst row. h fits in L2 (51 MB).
// ---------------------------------------------------------------------------
__global__ __launch_bounds__(256) void aggregate_kernel(
    const float* __restrict__ P, const int* __restrict__ src,
    const int* __restrict__ dst, float* __restrict__ AGG, int E) {
  const int e    = (blockIdx.x * 256 + threadIdx.x) >> 5;
  const int lane = threadIdx.x & 31;
  if (e >= E) return;
  const int s = src[e];
  const int d = dst[e];
  const float4 v = reinterpret_cast<const float4*>(P + (size_t)s * H_DIM)[lane];
  float* a = AGG + (size_t)d * H_DIM + lane * 4;
  atomic_add_f32(a + 0, v.x);
  atomic_add_f32(a + 1, v.y);
  atomic_add_f32(a + 2, v.z);
  atomic_add_f32(a + 3, v.w);
}

// ---------------------------------------------------------------------------
// h = relu(agg*nd + b) written IN PLACE over AGG; HMAX = running JK max.
// ---------------------------------------------------------------------------
__global__ __launch_bounds__(256) void epilogue_kernel(
    float* __restrict__ AGG, const float* __restrict__ nd,
    const float* __restrict__ bias, float* __restrict__ HMAX, int N, int first) {
  int idx   = blockIdx.x * 256 + threadIdx.x;        // float4 index
  int total = N * (H_DIM / 4);
  if (idx >= total) return;
  int   row = idx >> 5;            // 32 float4 per row
  int   c4  = (idx & 31) * 4;
  float s   = nd[row];
  float4 v  = reinterpret_cast<float4*>(AGG)[idx];
  float4 b  = *reinterpret_cast<const float4*>(bias + c4);
  v.x = fmaxf(fmaf(v.x, s, b.x), 0.f);
  v.y = fmaxf(fmaf(v.y, s, b.y), 0.f);
  v.z = fmaxf(fmaf(v.z, s, b.z), 0.f);
  v.w = fmaxf(fmaf(v.w, s, b.w), 0.f);
  reinterpret_cast<float4*>(AGG)[idx] = v;
  float4* hm = reinterpret_cast<float4*>(HMAX) + idx;
  if (first) {
    *hm = v;
  } else {
    float4 m = *hm;
    m.x = fmaxf(m.x, v.x);
    m.y = fmaxf(m.y, v.y);
    m.z = fmaxf(m.z, v.z);
    m.w = fmaxf(m.w, v.w);
    *hm = m;
  }
}

// ---------------------------------------------------------------------------
extern "C" void kernel_launch(void* const* d_in, const int* in_sizes, int n_in,
                              void* d_out, int out_size, void* d_ws, size_t ws_size,
                              hipStream_t stream) {
  (void)n_in; (void)out_size; (void)ws_size;
  const float* x    = (const float*)d_in[0];
  const int*   src  = (const int*)d_in[1];
  const int*   dst  = (const int*)d_in[2];
  const float* W0   = (const float*)d_in[3];
  const float* b0   = (const float*)d_in[4];
  const float* Wmid = (const float*)d_in[5];
  const float* bmid = (const float*)d_in[6];
  const float* Wl   = (const float*)d_in[7];
  const float* bl   = (const float*)d_in[8];

  const int F = 256, Hd = H_DIM, L = 6, C = 40;
  const int N = in_sizes[0] / F;
  const int E = in_sizes[1];

  // workspace: P | AGG (doubles as h, in place) | HMAX | odeg | ideg | ns | nd
  //            | Wt0 (128x256) | WtMid (5 x 128x128)
  float* P     = (float*)d_ws;
  float* AGG   = P + (size_t)N * Hd;
  float* HMAX  = AGG + (size_t)N * Hd;
  float* odeg  = HMAX + (size_t)N * Hd;
  float* ideg  = odeg + N;
  float* ns    = ideg + N;
  float* nd    = ns + N;
  float* Wt0   = nd + N;
  float* WtMid = Wt0 + (size_t)F * Hd;

  // weight transposes (tiny, once per launch)
  transpose_kernel<<<(F * Hd + 255) / 256, 256, 0, stream>>>(W0, Wt0, F, Hd);
  for (int l = 0; l < L - 1; ++l)
    transpose_kernel<<<(Hd * Hd + 255) / 256, 256, 0, stream>>>(
        Wmid + (size_t)l * Hd * Hd, WtMid + (size_t)l * Hd * Hd, Hd, Hd);

  hipMemsetAsync(odeg, 0, (size_t)2 * N * sizeof(float), stream);
  degree_kernel<<<(E + 255) / 256, 256, 0, stream>>>(src, dst, odeg, ideg, E);
  norm_kernel<<<(N + 255) / 256, 256, 0, stream>>>(odeg, ideg, ns, nd, N);

  constexpr int MT = 4;                      // 64 rows per block
  const int gemmBlocks = (N + MT * 16 - 1) / (MT * 16);
  const int tiles16    = (N + 15) / 16;
  const int aggBlocks  = (E * 32 + 255) / 256;   // one wave per edge
  const int epiBlocks  = (N * (Hd / 4) + 255) / 256;

  // --- layer 0: x[N,256] ---
  gemm_scale_kernel<256, MT><<<gemmBlocks, 256, 0, stream>>>(x, Wt0, ns, P, N);
  hipMemsetAsync(AGG, 0, (size_t)N * Hd * sizeof(float), stream);
  aggregate_kernel<<<aggBlocks, 256, 0, stream>>>(P, src, dst, AGG, E);
  epilogue_kernel<<<epiBlocks, 256, 0, stream>>>(AGG, nd, b0, HMAX, N, 1);

  // --- layers 1..5: h lives in AGG, projected into P, re-aggregated into AGG ---
  for (int l = 0; l < L - 1; ++l) {
    gemm_scale_kernel<128, MT><<<gemmBlocks, 256, 0, stream>>>(
        AGG, WtMid + (size_t)l * Hd * Hd, ns, P, N);
    hipMemsetAsync(AGG, 0, (size_t)N * Hd * sizeof(float), stream);
    aggregate_kernel<<<aggBlocks, 256, 0, stream>>>(P, src, dst, AGG, E);
    epilogue_kernel<<<epiBlocks, 256, 0, stream>>>(
        AGG, nd, bmid + (size_t)l * Hd, HMAX, N, 0);
  }

  // --- JK classifier ---
  gemm_final_kernel<<<tiles16, 96, 0, stream>>>(HMAX, Wl, bl, (float*)d_out, N, C);
}